// IntersectConcatAttention_40484361732519
// MI455X (gfx1250) — compile-verified
//
#include <hip/hip_runtime.h>
#include <math.h>

// ---------------- types ----------------
typedef __attribute__((ext_vector_type(2))) float v2f;
typedef __attribute__((ext_vector_type(8))) float v8f;

#define QD   512
#define SD   64
#define NCTX 32
#define NHEAD 4
#define EPS  1e-6f
#define SLOPE 0.01f

// LDS byte-offset of a __shared__ pointer: flat LDS aperture keeps the
// offset in the low 32 bits.
__device__ __forceinline__ unsigned lds_off(const void* p) {
    return (unsigned)(unsigned long long)p;
}

// async copy 16B global -> LDS (per lane), tracked by ASYNCcnt
__device__ __forceinline__ void async_b128(unsigned lds, const void* gptr) {
    asm volatile("global_load_async_to_lds_b128 %0, %1, off"
                 :: "v"(lds), "v"(gptr) : "memory");
}
__device__ __forceinline__ void wait_async0() {
    asm volatile("s_wait_asynccnt 0x0" ::: "memory");
}
__device__ __forceinline__ void wait_async6() {
    asm volatile("s_wait_asynccnt 0x6" ::: "memory");
}

// =====================================================================
// Kernel 1: per-batch attention + sigmoid + residual + pre-LayerNorm
// one block (256 threads) per batch element.
// key_embeds tile (64KB) is async-copied to LDS once, overlapped with
// the query-logit phase; both later passes read it from LDS.
// =====================================================================
#define KES 516   // ke_s row stride (floats): 516%64==4 -> conflict-free

__global__ __launch_bounds__(256) void attn_kernel(
    const float* __restrict__ ke,    // [B, 32, 512]
    const float* __restrict__ spa,   // [B, 32, 64]
    const float* __restrict__ q,     // [B, 512]
    const float* __restrict__ av,    // [1088, 4] row-major
    const float* __restrict__ ln_g,  // [512]
    const float* __restrict__ ln_b,  // [512]
    float* __restrict__ pre)         // [B, 512] out
{
    __shared__ float ke_s[NCTX * KES];          // 66048 B
    __shared__ float s_red[40];
    __shared__ float s_qlogit[NHEAD];
    __shared__ float s_logit[NHEAD][NCTX];
    __shared__ float s_ws[NCTX];

    const int b    = blockIdx.x;
    const int tid  = threadIdx.x;
    const int lane = tid & 31;
    const int wave = tid >> 5;

    const float* kbase = ke + (size_t)b * NCTX * QD;

    // ---- issue async ke tile: 4096 float4 over 256 threads ------------
    for (int it = 0; it < 16; ++it) {
        int f  = it * 256 + tid;        // float4 index
        int r  = f >> 7;                // row 0..31 (128 float4 per row)
        int c4 = f & 127;
        async_b128(lds_off(&ke_s[r * KES + c4 * 4]),
                   kbase + (size_t)r * QD + c4 * 4);
    }

    const float* qrow = q + (size_t)b * QD;

    // ---- phase 1 (overlaps async): q_logit[k] = sum_d q[d]*av[d][k] ----
    float q0 = qrow[tid];
    float q1 = qrow[tid + 256];
    const float4* av4 = (const float4*)av;
    float4 a0 = av4[tid];
    float4 a1 = av4[tid + 256];
    float p0 = q0 * a0.x + q1 * a1.x;
    float p1 = q0 * a0.y + q1 * a1.y;
    float p2 = q0 * a0.z + q1 * a1.z;
    float p3 = q0 * a0.w + q1 * a1.w;
    for (int off = 16; off > 0; off >>= 1) {
        p0 += __shfl_xor(p0, off, 32);
        p1 += __shfl_xor(p1, off, 32);
        p2 += __shfl_xor(p2, off, 32);
        p3 += __shfl_xor(p3, off, 32);
    }
    if (lane == 0) {
        s_red[wave * 4 + 0] = p0;
        s_red[wave * 4 + 1] = p1;
        s_red[wave * 4 + 2] = p2;
        s_red[wave * 4 + 3] = p3;
    }
    __syncthreads();
    if (tid < NHEAD) {
        float s = 0.f;
        for (int w = 0; w < 8; ++w) s += s_red[w * 4 + tid];
        s_qlogit[tid] = s;
    }
    wait_async0();          // ke tile landed (own wave's part)
    __syncthreads();        // ... and everyone else's

    // ---- phase 2: logits[n][k] over key(LDS) + spa(global) dims --------
    if (tid < NCTX * NHEAD) {
        const int n = tid >> 2;
        const int k = tid & 3;
        const float* srow = spa + ((size_t)b * NCTX + n) * SD;
        const float* avk  = av + QD * NHEAD + k;          // av[(512+d)*4+k]
        const float* avs  = av + 2 * QD * NHEAD + k;      // av[(1024+d)*4+k]
        const float* krow = &ke_s[n * KES];
        float acc = s_qlogit[k];
        for (int d = 0; d < QD; d += 4) {
            float4 kv = *(const float4*)(krow + d);
            acc += kv.x * avk[4 * d]      + kv.y * avk[4 * d + 4] +
                   kv.z * avk[4 * d + 8]  + kv.w * avk[4 * d + 12];
        }
        for (int d = 0; d < SD; d += 4) {
            float4 sv = *(const float4*)(srow + d);
            acc += sv.x * avs[4 * d]      + sv.y * avs[4 * d + 4] +
                   sv.z * avs[4 * d + 8]  + sv.w * avs[4 * d + 12];
        }
        s_logit[k][n] = acc;
    }
    __syncthreads();

    // ---- phase 3: leaky-relu + softmax over n, average heads -----------
    if (tid < NCTX) {
        const int n = tid;
        float wsum = 0.f;
        for (int k = 0; k < NHEAD; ++k) {
            float x = s_logit[k][n];
            x = (x > 0.f) ? x : SLOPE * x;
            float mx = x;
            for (int off = 16; off > 0; off >>= 1)
                mx = fmaxf(mx, __shfl_xor(mx, off, 32));
            float e = __expf(x - mx);
            float s = e;
            for (int off = 16; off > 0; off >>= 1)
                s += __shfl_xor(s, off, 32);
            wsum += e / s;
        }
        s_ws[n] = 0.25f * wsum;
    }
    __syncthreads();

    // ---- phase 4: combined[d] = sum_n ws[n]*ke_s[n][d]; sigmoid; +q ----
    float c0 = 0.f, c1 = 0.f;
    for (int n = 0; n < NCTX; ++n) {
        float w = s_ws[n];
        c0 += w * ke_s[n * KES + tid];
        c1 += w * ke_s[n * KES + tid + 256];
    }
    c0 = 1.f / (1.f + __expf(-c0)) + q0;
    c1 = 1.f / (1.f + __expf(-c1)) + q1;

    // ---- phase 5: LayerNorm over 512 -----------------------------------
    float sum = c0 + c1;
    float sq  = c0 * c0 + c1 * c1;
    for (int off = 16; off > 0; off >>= 1) {
        sum += __shfl_xor(sum, off, 32);
        sq  += __shfl_xor(sq,  off, 32);
    }
    __syncthreads();   // s_red reuse
    if (lane == 0) { s_red[wave] = sum; s_red[8 + wave] = sq; }
    __syncthreads();
    if (tid == 0) {
        float S = 0.f, Q = 0.f;
        for (int w = 0; w < 8; ++w) { S += s_red[w]; Q += s_red[8 + w]; }
        float m = S * (1.f / QD);
        float v = Q * (1.f / QD) - m * m;
        s_red[16] = m;
        s_red[17] = rsqrtf(v + EPS);
    }
    __syncthreads();
    float m = s_red[16], r = s_red[17];
    float* orow = pre + (size_t)b * QD;
    orow[tid]       = ln_g[tid]       * (c0 - m) * r + ln_b[tid];
    orow[tid + 256] = ln_g[tid + 256] * (c1 - m) * r + ln_b[tid + 256];
}

// =====================================================================
// Kernel 2: tmp = pre @ post_w^T + post_b + pre   (WMMA f32 16x16x4)
// block tile 128x64, 8 waves (4x2), wave tile 32x32 (2x2 accumulators)
// double-buffered LDS fed by async-to-LDS copies (ASYNCcnt)
// =====================================================================
#define BM 128
#define BN 64
#define BK 32
#define NT (QD / BK)   // 16 k-tiles
#define KSTR 36        // 36*4B = 144B: 16B-aligned rows, conflict-free b64

__global__ __launch_bounds__(256) void gemm_kernel(
    const float* __restrict__ A,     // pre   [B, 512]
    const float* __restrict__ W,     // post_w[512, 512] (j, i)
    const float* __restrict__ bias,  // post_b[512]
    float* __restrict__ out)         // tmp   [B, 512]
{
    __shared__ float As[2][BM * KSTR];   // 2 x 18432 B
    __shared__ float Bs[2][BN * KSTR];   // 2 x  9216 B

    const int tid  = threadIdx.x;
    const int lane = tid & 31;
    const int wave = tid >> 5;
    const int wm   = wave >> 1;       // 0..3
    const int wn   = wave & 1;        // 0..1
    const int bm0  = blockIdx.x * BM;
    const int bn0  = blockIdx.y * BN;

    const int ml = lane & 15;         // M (A) / N (B) within 16
    const int kh = (lane >> 4) * 2;   // K sub-offset: lanes 0-15 -> 0, 16-31 -> 2

    v8f acc[2][2] = {};

    // per-thread staging coordinates (A: 4 x b128, B: 2 x b128)
    const int ra[4] = { (tid >> 3), (256 + tid) >> 3, (512 + tid) >> 3, (768 + tid) >> 3 };
    const int rb[2] = { (tid >> 3), (256 + tid) >> 3 };
    const int cc = (tid & 7) * 4;

    // issue tile 0 into buffer 0
    {
        const int k0 = 0;
        for (int it = 0; it < 4; ++it)
            async_b128(lds_off(&As[0][ra[it] * KSTR + cc]),
                       A + (size_t)(bm0 + ra[it]) * QD + k0 + cc);
        for (int it = 0; it < 2; ++it)
            async_b128(lds_off(&Bs[0][rb[it] * KSTR + cc]),
                       W + (size_t)(bn0 + rb[it]) * QD + k0 + cc);
    }

    for (int t = 0; t < NT; ++t) {
        const int cur = t & 1;
        if (t + 1 < NT) {
            const int nxt = (t + 1) & 1;
            const int k1  = (t + 1) * BK;
            for (int it = 0; it < 4; ++it)
                async_b128(lds_off(&As[nxt][ra[it] * KSTR + cc]),
                           A + (size_t)(bm0 + ra[it]) * QD + k1 + cc);
            for (int it = 0; it < 2; ++it)
                async_b128(lds_off(&Bs[nxt][rb[it] * KSTR + cc]),
                           W + (size_t)(bn0 + rb[it]) * QD + k1 + cc);
            wait_async6();        // tile t (older 6 ops) has landed
        } else {
            wait_async0();
        }
        __syncthreads();

        const float* Ab = &As[cur][0];
        const float* Bb = &Bs[cur][0];
        for (int kk = 0; kk < BK; kk += 4) {
            v2f a0 = *(const v2f*)(Ab + (wm * 32 +      ml) * KSTR + kk + kh);
            v2f a1 = *(const v2f*)(Ab + (wm * 32 + 16 + ml) * KSTR + kk + kh);
            v2f b0 = *(const v2f*)(Bb + (wn * 32 +      ml) * KSTR + kk + kh);
            v2f b1 = *(const v2f*)(Bb + (wn * 32 + 16 + ml) * KSTR + kk + kh);
            acc[0][0] = __builtin_amdgcn_wmma_f32_16x16x4_f32(
                false, a0, false, b0, (short)0, acc[0][0], false, false);
            acc[0][1] = __builtin_amdgcn_wmma_f32_16x16x4_f32(
                false, a0, false, b1, (short)0, acc[0][1], false, false);
            acc[1][0] = __builtin_amdgcn_wmma_f32_16x16x4_f32(
                false, a1, false, b0, (short)0, acc[1][0], false, false);
            acc[1][1] = __builtin_amdgcn_wmma_f32_16x16x4_f32(
                false, a1, false, b1, (short)0, acc[1][1], false, false);
        }
        __syncthreads();   // all waves done with buffer `cur` before reuse
    }

    // epilogue: + bias + residual(pre), write tmp
    const int hh = lane >> 4;
    for (int i = 0; i < 2; ++i) {
        for (int j = 0; j < 2; ++j) {
            int col = bn0 + wn * 32 + j * 16 + ml;
            float bc = bias[col];
            for (int v = 0; v < 8; ++v) {
                int row = bm0 + wm * 32 + i * 16 + v + 8 * hh;
                size_t idx = (size_t)row * QD + col;
                out[idx] = acc[i][j][v] + bc + A[idx];
            }
        }
    }
}

// =====================================================================
// Kernel 3: final row LayerNorm -> d_out
// =====================================================================
__global__ __launch_bounds__(256) void ln_kernel(
    const float* __restrict__ x,     // tmp [B, 512]
    const float* __restrict__ g,
    const float* __restrict__ beta,
    float* __restrict__ out)
{
    __shared__ float s_red[20];
    const int b = blockIdx.x;
    const int tid = threadIdx.x;
    const int lane = tid & 31;
    const int wave = tid >> 5;

    const float* row = x + (size_t)b * QD;
    float c0 = row[tid];
    float c1 = row[tid + 256];
    float sum = c0 + c1;
    float sq  = c0 * c0 + c1 * c1;
    for (int off = 16; off > 0; off >>= 1) {
        sum += __shfl_xor(sum, off, 32);
        sq  += __shfl_xor(sq,  off, 32);
    }
    if (lane == 0) { s_red[wave] = sum; s_red[8 + wave] = sq; }
    __syncthreads();
    if (tid == 0) {
        float S = 0.f, Q = 0.f;
        for (int w = 0; w < 8; ++w) { S += s_red[w]; Q += s_red[8 + w]; }
        float m = S * (1.f / QD);
        float v = Q * (1.f / QD) - m * m;
        s_red[16] = m;
        s_red[17] = rsqrtf(v + EPS);
    }
    __syncthreads();
    float m = s_red[16], r = s_red[17];
    float* orow = out + (size_t)b * QD;
    orow[tid]       = g[tid]       * (c0 - m) * r + beta[tid];
    orow[tid + 256] = g[tid + 256] * (c1 - m) * r + beta[tid + 256];
}

// =====================================================================
extern "C" void kernel_launch(void* const* d_in, const int* in_sizes, int n_in,
                              void* d_out, int out_size, void* d_ws, size_t ws_size,
                              hipStream_t stream)
{
    const float* ke   = (const float*)d_in[0];   // [B,32,512]
    const float* spa  = (const float*)d_in[1];   // [B,32,64]
    const float* q    = (const float*)d_in[2];   // [B,512]
    const float* av   = (const float*)d_in[3];   // [1088,4]
    const float* pw   = (const float*)d_in[4];   // [512,512]
    const float* pb   = (const float*)d_in[5];   // [512]
    const float* g1   = (const float*)d_in[6];
    const float* b1   = (const float*)d_in[7];
    const float* g2   = (const float*)d_in[8];
    const float* b2   = (const float*)d_in[9];
    float* out = (float*)d_out;

    const int B = in_sizes[2] / QD;              // 4096

    float* pre = (float*)d_ws;                   // [B,512]
    float* tmp = pre + (size_t)B * QD;           // [B,512]

    attn_kernel<<<B, 256, 0, stream>>>(ke, spa, q, av, g1, b1, pre);
    gemm_kernel<<<dim3(B / BM, QD / BN), 256, 0, stream>>>(pre, pw, pb, tmp);
    ln_kernel<<<B, 256, 0, stream>>>(tmp, g2, b2, out);
}